// pFedGPIPCompute_71468255805516
// MI455X (gfx1250) — compile-verified
//
#include <hip/hip_runtime.h>
#include <hip/hip_bf16.h>

// Sizes fixed by the reference
#define NN 2048   // data points
#define MM 512    // inducing points
#define DD 256    // feature dim
#define ND_ 20    // parallel chains
#define SS 10     // scan steps
#define JITTER 1e-4f

typedef __attribute__((ext_vector_type(2))) float v2f;
typedef __attribute__((ext_vector_type(8))) float v8f;

__device__ __forceinline__ float pg_mean1(float c) {
  float ac = __builtin_fabsf(c);
  if (ac < 1e-6f) return 0.25f;
  return tanhf(0.5f * c) / (2.0f * c);
}

__device__ __forceinline__ v8f wmma_f32(v2f a, v2f b, v8f c) {
  return __builtin_amdgcn_wmma_f32_16x16x4_f32(false, a, false, b,
                                               (short)0, c, false, false);
}

// ---------------------------------------------------------------- row norms
__global__ void norms_kernel(const float* __restrict__ X,
                             const float* __restrict__ Xbar,
                             float* __restrict__ norms) {
  int i = blockIdx.x * blockDim.x + threadIdx.x;
  if (i >= MM + NN) return;
  const float* row = (i < MM) ? (Xbar + (size_t)i * DD) : (X + (size_t)(i - MM) * DD);
  float s = 0.0f;
  for (int k = 0; k < DD; ++k) s += row[k] * row[k];
  norms[i] = s;
}

// --------------------------------------------- WMMA gram + RBF transform
// Kout[i][j] = os * exp(-0.5*(|xi|^2+|xj|^2-2 xi.xj)/ls^2), i<2560, j<512
__global__ void gram_rbf_kernel(const float* __restrict__ X,
                                const float* __restrict__ Xbar,
                                const float* __restrict__ norms,
                                const float* __restrict__ ls_p,
                                const float* __restrict__ os_p,
                                float* __restrict__ Kout) {
  const int j0 = blockIdx.x * 16;      // 0..511
  const int i0 = blockIdx.y * 16;      // 0..2559
  const int lane = threadIdx.x;        // 32 threads = 1 wave
  const int grp = lane >> 4;           // lane group selects K-pair
  const int lr  = lane & 15;
  const int ia = i0 + lr;
  const float* arow = (ia < MM) ? (Xbar + (size_t)ia * DD) : (X + (size_t)(ia - MM) * DD);
  const float* brow = Xbar + (size_t)(j0 + lr) * DD;   // j always < 512
  v8f acc = {};
  for (int k = 0; k < DD; k += 4) {
    v2f a = *(const v2f*)(arow + k + 2 * grp);   // A[M=lr][K=k+2g .. +1]
    v2f b = *(const v2f*)(brow + k + 2 * grp);   // B[K=k+2g .. +1][N=lr]
    acc = wmma_f32(a, b, acc);
  }
  const float ls = ls_p[0], os_ = os_p[0];
  const float sc = -0.5f / (ls * ls);
  const float nj = norms[j0 + lr];
  const int col = j0 + lr;
#pragma unroll
  for (int v = 0; v < 8; ++v) {
    int row = i0 + v + 8 * grp;
    float d2 = norms[row] + nj - 2.0f * acc[v];
    d2 = fmaxf(d2, 0.0f);
    Kout[(size_t)row * MM + col] = os_ * __expf(sc * d2);
  }
}

// ------------------------------------------------ WMMA batched Q build
// Q[d][a][b] = Kmm[a][b] + sum_n Knm[n][a]*lam[d][n]*Knm[n][b] (+jitter diag)
// 2x2 register blocking: each wave produces a 32x32 output tile (4 WMMA tiles)
__global__ void qgemm_kernel(const float* __restrict__ Knm,
                             const float* __restrict__ Kmm,
                             const float* __restrict__ lam,
                             float* __restrict__ Q) {
  const int b0 = blockIdx.x * 32;
  const int a0 = blockIdx.y * 32;
  const int d  = blockIdx.z;
  const int lane = threadIdx.x, grp = lane >> 4, lr = lane & 15;
  const float* lamd = lam + (size_t)d * NN;
  v8f acc00 = {}, acc01 = {}, acc10 = {}, acc11 = {};
  for (int n = 0; n < NN; n += 4) {
    const int n0 = n + 2 * grp;
    const float l0 = lamd[n0], l1 = lamd[n0 + 1];
    const float* r0 = Knm + (size_t)n0 * MM;
    const float* r1 = r0 + MM;
    // prefetch next K-panel rows (global_prefetch_b8)
    if (n + 4 < NN) {
      __builtin_prefetch(r0 + 4 * MM + a0 + lr, 0, 1);
      __builtin_prefetch(r0 + 4 * MM + b0 + lr, 0, 1);
    }
    v2f aA, aB, bA, bB;
    aA.x = r0[a0 + lr] * l0;        aA.y = r1[a0 + lr] * l1;
    aB.x = r0[a0 + 16 + lr] * l0;   aB.y = r1[a0 + 16 + lr] * l1;
    bA.x = r0[b0 + lr];             bA.y = r1[b0 + lr];
    bB.x = r0[b0 + 16 + lr];        bB.y = r1[b0 + 16 + lr];
    acc00 = wmma_f32(aA, bA, acc00);
    acc01 = wmma_f32(aA, bB, acc01);
    acc10 = wmma_f32(aB, bA, acc10);
    acc11 = wmma_f32(aB, bB, acc11);
  }
  float* Qd = Q + (size_t)d * MM * MM;
#pragma unroll
  for (int v = 0; v < 8; ++v) {
    const int r_lo = a0 + v + 8 * grp;
    const int r_hi = r_lo + 16;
    const int c_lo = b0 + lr;
    const int c_hi = c_lo + 16;
    float q;
    q = acc00[v] + Kmm[(size_t)r_lo * MM + c_lo] + ((r_lo == c_lo) ? JITTER : 0.0f);
    Qd[(size_t)r_lo * MM + c_lo] = q;
    q = acc01[v] + Kmm[(size_t)r_lo * MM + c_hi] + ((r_lo == c_hi) ? JITTER : 0.0f);
    Qd[(size_t)r_lo * MM + c_hi] = q;
    q = acc10[v] + Kmm[(size_t)r_hi * MM + c_lo] + ((r_hi == c_lo) ? JITTER : 0.0f);
    Qd[(size_t)r_hi * MM + c_lo] = q;
    q = acc11[v] + Kmm[(size_t)r_hi * MM + c_hi] + ((r_hi == c_hi) ? JITTER : 0.0f);
    Qd[(size_t)r_hi * MM + c_hi] = q;
  }
}

// --------------------------------------------------- Kmm + jitter -> L0
__global__ void copyjit_kernel(const float* __restrict__ Kmm, float* __restrict__ L0) {
  int idx = blockIdx.x * blockDim.x + threadIdx.x;
  if (idx >= MM * MM) return;
  int rr = idx >> 9, cc = idx & (MM - 1);
  L0[idx] = Kmm[idx] + ((rr == cc) ? JITTER : 0.0f);
}

// --------------------------------------- in-place Cholesky (lower), batched
// Right-looking, row-cyclic trailing update; pivot column cached in LDS.
__global__ void cholesky_kernel(float* __restrict__ A, int batch_stride) {
  float* Mtx = A + (size_t)blockIdx.x * batch_stride;
  const int n = MM;
  const int tid = threadIdx.x, nt = blockDim.x;
  __shared__ float colk[MM];
  for (int k = 0; k < n; ++k) {
    if (tid == 0) Mtx[(size_t)k * n + k] = sqrtf(fmaxf(Mtx[(size_t)k * n + k], 1e-20f));
    __syncthreads();
    const float dk = Mtx[(size_t)k * n + k];
    for (int i = k + 1 + tid; i < n; i += nt) {
      float v = Mtx[(size_t)i * n + k] / dk;
      Mtx[(size_t)i * n + k] = v;
      colk[i] = v;
    }
    __syncthreads();
    for (int i = k + 1 + tid; i < n; i += nt) {
      const float lik = colk[i];
      float* row = Mtx + (size_t)i * n;
      for (int j = k + 1; j <= i; ++j) row[j] -= lik * colk[j];
    }
    __syncthreads();
  }
}

// ------------------------- wave-per-column forward solve  z = L^-1 * Kmn[:,n]
// Optionally stores z (Z0 path) and/or sum(z^2) (per-step sd path).
__global__ void trsv_fwd_kernel(const float* __restrict__ L, int Lstride,
                                const float* __restrict__ Knm,
                                float* __restrict__ z_out,
                                float* __restrict__ sumsq_out) {
  const int n = blockIdx.x;            // rhs column 0..2047
  const int d = blockIdx.y;
  const int lane = threadIdx.x;        // 32 threads, one wave
  const float* Ld = L + (size_t)d * Lstride;
  __shared__ float x[MM];
  for (int m = lane; m < MM; m += 32) x[m] = Knm[(size_t)n * MM + m]; // Kmn[m,n]
  __syncthreads();
  for (int i = 0; i < MM; ++i) {
    const float* Lrow = Ld + (size_t)i * MM;
    float s = 0.0f;
    for (int j = lane; j < i; j += 32) s += Lrow[j] * x[j];
#pragma unroll
    for (int off = 16; off > 0; off >>= 1) s += __shfl_xor(s, off, 32);
    float xi = (x[i] - s) / Lrow[i];
    __syncthreads();
    if (lane == 0) x[i] = xi;
    __syncthreads();
  }
  if (z_out != nullptr) {
    for (int m = lane; m < MM; m += 32) z_out[(size_t)m * NN + n] = x[m];
  }
  if (sumsq_out != nullptr) {
    float ss = 0.0f;
    for (int m = lane; m < MM; m += 32) ss += x[m] * x[m];
#pragma unroll
    for (int off = 16; off > 0; off >>= 1) ss += __shfl_xor(ss, off, 32);
    if (lane == 0) sumsq_out[(size_t)d * NN + n] = ss;
  }
}

// ------------------------------------- init: prior, f0, omega0 from Z0
__global__ void init_kernel(const float* __restrict__ Z0,
                            const float* __restrict__ eps_u,
                            const float* __restrict__ eps_f0,
                            const float* __restrict__ os_p,
                            float* __restrict__ prior,
                            float* __restrict__ omega) {
  int n = blockIdx.x * blockDim.x + threadIdx.x;
  if (n >= NN) return;
  float acc[ND_];
#pragma unroll
  for (int j = 0; j < ND_; ++j) acc[j] = 0.0f;
  float ss = 0.0f;
  for (int m = 0; m < MM; ++m) {
    float z = Z0[(size_t)m * NN + n];
    ss += z * z;
    const float* eu = eps_u + (size_t)m * ND_;
#pragma unroll
    for (int j = 0; j < ND_; ++j) acc[j] += z * eu[j];
  }
  float pr = os_p[0] - ss;     // knn_diag == outputscale for RBF
  prior[n] = pr;
  float sp = sqrtf(fmaxf(pr, 0.0f));
  float e0 = eps_f0[n];
#pragma unroll
  for (int j = 0; j < ND_; ++j)
    omega[(size_t)j * NN + n] = pg_mean1(acc[j] + sp * e0);
}

// ------------------------------------------------- lam / vec per (d,n)
__global__ void lamvec_kernel(const float* __restrict__ omega,
                              const float* __restrict__ prior,
                              const int* __restrict__ Y,
                              float* __restrict__ lam,
                              float* __restrict__ vec) {
  int idx = blockIdx.x * blockDim.x + threadIdx.x;
  if (idx >= ND_ * NN) return;
  int n = idx & (NN - 1);
  float om = omega[idx];
  float li = 1.0f / (1.0f / om + prior[n]);
  float kap = (Y[n] == 0) ? 0.5f : -0.5f;
  lam[idx] = li;
  vec[idx] = li * (kap / om);
}

// --------------------------------------------------- r[d] = Kmn @ vec[d]
__global__ void kmnvec_kernel(const float* __restrict__ Knm,
                              const float* __restrict__ vec,
                              float* __restrict__ r) {
  int m = threadIdx.x;                 // 512 threads
  int d = blockIdx.x;
  const float* vd = vec + (size_t)d * NN;
  float s = 0.0f;
  for (int n = 0; n < NN; ++n) s += Knm[(size_t)n * MM + m] * vd[n];
  r[(size_t)d * MM + m] = s;
}

// ---------------------------------- fwd+bwd solve per d: w = Q^-1 r (wave)
__global__ void small_solve_kernel(const float* __restrict__ LQ,
                                   const float* __restrict__ r,
                                   float* __restrict__ w) {
  const int d = blockIdx.x;
  const int lane = threadIdx.x;
  const float* Ld = LQ + (size_t)d * MM * MM;
  __shared__ float x[MM];
  for (int m = lane; m < MM; m += 32) x[m] = r[(size_t)d * MM + m];
  __syncthreads();
  for (int i = 0; i < MM; ++i) {                 // L y = r
    const float* Lr = Ld + (size_t)i * MM;
    float s = 0.0f;
    for (int j = lane; j < i; j += 32) s += Lr[j] * x[j];
#pragma unroll
    for (int off = 16; off > 0; off >>= 1) s += __shfl_xor(s, off, 32);
    float xi = (x[i] - s) / Lr[i];
    __syncthreads();
    if (lane == 0) x[i] = xi;
    __syncthreads();
  }
  for (int i = MM - 1; i >= 0; --i) {            // L^T w = y
    float s = 0.0f;
    for (int j = i + 1 + lane; j < MM; j += 32) s += Ld[(size_t)j * MM + i] * x[j];
#pragma unroll
    for (int off = 16; off > 0; off >>= 1) s += __shfl_xor(s, off, 32);
    float xi = (x[i] - s) / Ld[(size_t)i * MM + i];
    __syncthreads();
    if (lane == 0) x[i] = xi;
    __syncthreads();
  }
  for (int m = lane; m < MM; m += 32) w[(size_t)d * MM + m] = x[m];
}

// ---------------------------- finalize step: mu, sd, f, omega (+out last)
__global__ void finalize_kernel(const float* __restrict__ Knm,
                                const float* __restrict__ w,
                                const float* __restrict__ prior,
                                const float* __restrict__ sumZ2,
                                const float* __restrict__ eps_s,
                                float* __restrict__ omega,
                                float* __restrict__ out, int is_last) {
  int n = blockIdx.x * blockDim.x + threadIdx.x;
  int d = blockIdx.y;
  if (n >= NN) return;
  const float* wd = w + (size_t)d * MM;
  const float* krow = Knm + (size_t)n * MM;
  float mu = 0.0f;
  for (int m = 0; m < MM; ++m) mu += krow[m] * wd[m];
  float sd = prior[n] + sumZ2[(size_t)d * NN + n];
  float f = mu + sqrtf(fmaxf(sd, 1e-10f)) * eps_s[(size_t)d * NN + n];
  omega[(size_t)d * NN + n] = pg_mean1(f);
  if (is_last) out[(size_t)d * NN + n] = f;
}

extern "C" void kernel_launch(void* const* d_in, const int* in_sizes, int n_in,
                              void* d_out, int out_size, void* d_ws, size_t ws_size,
                              hipStream_t stream) {
  const float* X      = (const float*)d_in[0];   // (2048,256)
  const float* Xbar   = (const float*)d_in[1];   // (512,256)
  const float* ls     = (const float*)d_in[2];   // (1,)
  const float* os_    = (const float*)d_in[3];   // (1,)
  const float* eps_u  = (const float*)d_in[4];   // (512,20)
  const float* eps_f0 = (const float*)d_in[5];   // (2048,1)
  const float* eps_st = (const float*)d_in[6];   // (10,20,2048)
  const int*   Y      = (const int*)d_in[7];     // (2048,)
  float* out = (float*)d_out;                    // (20,2048)

  float* ws   = (float*)d_ws;
  float* norms = ws;                ws += (MM + NN);
  float* Kfull = ws;                ws += (size_t)(MM + NN) * MM;
  float* Kmm = Kfull;                                  // rows 0..511
  float* Knm = Kfull + (size_t)MM * MM;                // rows 512..2559 (n-major)
  float* L0    = ws;                ws += (size_t)MM * MM;
  float* Z0    = ws;                ws += (size_t)MM * NN;
  float* prior = ws;                ws += NN;
  float* omega = ws;                ws += (size_t)ND_ * NN;
  float* lam   = ws;                ws += (size_t)ND_ * NN;
  float* vec   = ws;                ws += (size_t)ND_ * NN;
  float* sumZ2 = ws;                ws += (size_t)ND_ * NN;
  float* Q     = ws;                ws += (size_t)ND_ * MM * MM;
  float* r     = ws;                ws += (size_t)ND_ * MM;
  float* w     = ws;                ws += (size_t)ND_ * MM;

  // ---- setup phase
  norms_kernel<<<(MM + NN + 255) / 256, 256, 0, stream>>>(X, Xbar, norms);
  gram_rbf_kernel<<<dim3(MM / 16, (MM + NN) / 16), 32, 0, stream>>>(
      X, Xbar, norms, ls, os_, Kfull);
  copyjit_kernel<<<(MM * MM + 255) / 256, 256, 0, stream>>>(Kmm, L0);
  cholesky_kernel<<<1, 512, 0, stream>>>(L0, 0);
  trsv_fwd_kernel<<<dim3(NN, 1), 32, 0, stream>>>(L0, 0, Knm, Z0, nullptr);
  init_kernel<<<(NN + 63) / 64, 64, 0, stream>>>(Z0, eps_u, eps_f0, os_, prior, omega);

  // ---- S Gibbs-style steps (sequential dependency through omega)
  for (int s = 0; s < SS; ++s) {
    lamvec_kernel<<<(ND_ * NN + 255) / 256, 256, 0, stream>>>(omega, prior, Y, lam, vec);
    qgemm_kernel<<<dim3(MM / 32, MM / 32, ND_), 32, 0, stream>>>(Knm, Kmm, lam, Q);
    cholesky_kernel<<<ND_, 512, 0, stream>>>(Q, MM * MM);
    trsv_fwd_kernel<<<dim3(NN, ND_), 32, 0, stream>>>(Q, MM * MM, Knm, nullptr, sumZ2);
    kmnvec_kernel<<<ND_, MM, 0, stream>>>(Knm, vec, r);
    small_solve_kernel<<<ND_, 32, 0, stream>>>(Q, r, w);
    finalize_kernel<<<dim3((NN + 255) / 256, ND_), 256, 0, stream>>>(
        Knm, w, prior, sumZ2, eps_st + (size_t)s * ND_ * NN, omega, out,
        (s == SS - 1) ? 1 : 0);
  }
}